// LabelSpreading_67035849556468
// MI455X (gfx1250) — compile-verified
//
#include <hip/hip_runtime.h>
#include <math.h>

// ---------------- problem constants (match reference setup_inputs) ----------
#define LS_GAMMA  20.0f
#define LS_FACTOR 0.99f

#define C_DIM   640          // channels
#define HW      49           // h*w
#define NW      5            // n_way
#define KS      5            // k_shot
#define S_TOT   25           // NW*KS support shots
#define MS      245          // support graph nodes NW*HW
#define MT      294          // total graph nodes MS+HW
#define NAUG    299          // MT + NW  (augmented [A | Y])
#define AUGW    304          // padded col stride of augmented matrix
#define AUGH    304          // padded row count (unconditional WMMA tile stores)
#define NTIL    19           // ceil(MT/16) tiles per dim

typedef __attribute__((ext_vector_type(2))) float v2f;
typedef __attribute__((ext_vector_type(8))) float v8f;

// ---------------------------------------------------------------------------
// Kernel 0: zero the shared pad column (re-done every launch: deterministic,
// no reliance on prior buffer state).
// ---------------------------------------------------------------------------
__global__ void k_zero(float* __restrict__ zcol) {
  zcol[blockIdx.x * blockDim.x + threadIdx.x] = 0.0f;
}

// ---------------------------------------------------------------------------
// Kernel 1: class prototypes (mean over shots) + L2 norm over channels.
// One block per (b, m) column, m = n*HW + j (class-major). Output column-major
// over c: sflat[(bb*MS + m)*C + ch].
// ---------------------------------------------------------------------------
__global__ void k_proto_norm(const float* __restrict__ sup,
                             float* __restrict__ sflat) {
  const int blk = blockIdx.x;
  const int bb  = blk / MS;
  const int m   = blk % MS;
  const int n   = m / HW;
  const int j   = m % HW;
  const int t   = threadIdx.x;            // 64 threads
  float v[C_DIM / 64];
  float ss = 0.0f;
  #pragma unroll
  for (int u = 0; u < C_DIM / 64; ++u) {
    const int ch = t + u * 64;
    const float* p = sup + (((size_t)bb * S_TOT + n * KS) * C_DIM + ch) * HW + j;
    float acc = 0.0f;
    #pragma unroll
    for (int sh = 0; sh < KS; ++sh) acc += p[(size_t)sh * C_DIM * HW];
    acc *= (1.0f / KS);
    v[u] = acc;
    ss += acc * acc;
  }
  __shared__ float red[64];
  red[t] = ss;
  __syncthreads();
  for (int o = 32; o > 0; o >>= 1) {
    if (t < o) red[t] += red[t + o];
    __syncthreads();
  }
  const float rn = rsqrtf(red[0] + 1e-12f);
  float* outp = sflat + ((size_t)bb * MS + m) * C_DIM;
  #pragma unroll
  for (int u = 0; u < C_DIM / 64; ++u) outp[t + u * 64] = v[u] * rn;
}

// ---------------------------------------------------------------------------
// Kernel 2: L2-normalize query features per (e, j) column over channels.
// query_xf is (b,q,c,hw) contiguous == (e,c,hw). Output qfn[(e*HW + j)*C + ch].
// ---------------------------------------------------------------------------
__global__ void k_query_norm(const float* __restrict__ qry,
                             float* __restrict__ qfn) {
  const int blk = blockIdx.x;              // nex*HW blocks
  const int e   = blk / HW;
  const int j   = blk % HW;
  const int t   = threadIdx.x;             // 64 threads
  float v[C_DIM / 64];
  float ss = 0.0f;
  #pragma unroll
  for (int u = 0; u < C_DIM / 64; ++u) {
    const int ch = t + u * 64;
    const float x = qry[((size_t)e * C_DIM + ch) * HW + j];
    v[u] = x;
    ss += x * x;
  }
  __shared__ float red[64];
  red[t] = ss;
  __syncthreads();
  for (int o = 32; o > 0; o >>= 1) {
    if (t < o) red[t] += red[t + o];
    __syncthreads();
  }
  const float rn = rsqrtf(red[0] + 1e-12f);
  float* outp = qfn + ((size_t)e * HW + j) * C_DIM;
  #pragma unroll
  for (int u = 0; u < C_DIM / 64; ++u) outp[t + u * 64] = v[u] * rn;
}

// ---------------------------------------------------------------------------
// Kernel 3: raw Gram W = G^T G via V_WMMA_F32_16X16X4_F32.
// G_e (640 x 294) = [sflat_b | qf_e], columns contiguous over c.
// Out-of-range columns (last tile) point at a shared zero column, so the
// K-loop is pure {2x global_load_b64 -> v_wmma} with no mask VALU ops.
// One wave (32 threads) per (example, 16x16 output tile); EXEC stays all-1s.
// f32 A-fragment layout (ISA 7.12.2): lane M = lane%16;
//   lanes 0-15:  VGPR0=K0, VGPR1=K1 ; lanes 16-31: VGPR0=K2, VGPR1=K3.
// Stores are unconditional into the 304x304 padded slab.
// ---------------------------------------------------------------------------
__global__ void k_gram_wmma(const float* __restrict__ sflat,
                            const float* __restrict__ qfn,
                            const float* __restrict__ zcol,
                            float* __restrict__ aug, int q) {
  const int blk  = blockIdx.x;
  const int e    = blk / (NTIL * NTIL);
  const int tile = blk % (NTIL * NTIL);
  const int tm   = tile / NTIL;
  const int tn   = tile % NTIL;
  const int bb   = e / q;
  const int lane = threadIdx.x;            // 32 threads = 1 wave
  const int mrow  = lane & 15;
  const int khalf = (lane >> 4) << 1;      // 0 or 2

  const int colA = tm * 16 + mrow;
  const int colB = tn * 16 + mrow;
  const float* pA = (colA >= MT) ? zcol
                  : (colA < MS)  ? (sflat + ((size_t)bb * MS + colA) * C_DIM)
                                 : (qfn + ((size_t)e * HW + (colA - MS)) * C_DIM);
  const float* pB = (colB >= MT) ? zcol
                  : (colB < MS)  ? (sflat + ((size_t)bb * MS + colB) * C_DIM)
                                 : (qfn + ((size_t)e * HW + (colB - MS)) * C_DIM);

  v8f acc = {};
  #pragma unroll 4
  for (int k0 = 0; k0 < C_DIM; k0 += 4) {
    const int k = k0 + khalf;               // 8B-aligned (k even, col base aligned)
    const v2f a = *(const v2f*)(pA + k);
    const v2f b = *(const v2f*)(pB + k);
    acc = __builtin_amdgcn_wmma_f32_16x16x4_f32(
        /*neg_a=*/false, a, /*neg_b=*/false, b,
        /*c_mod=*/(short)0, acc, /*reuse_a=*/false, /*reuse_b=*/false);
  }

  float* A = aug + (size_t)e * AUGH * AUGW;
  const int rbase = tm * 16 + ((lane >> 4) << 3);  // +0 or +8
  const int col   = tn * 16 + mrow;
  #pragma unroll
  for (int r = 0; r < 8; ++r) {
    A[(size_t)(rbase + r) * AUGW + col] = acc[r];   // always in padded bounds
  }
}

// ---------------------------------------------------------------------------
// Kernel 4: in-place affinity transform + label seeding.
// Phase 1: dinv_i = rsqrt( sum_j exp(20 g_ij), j!=i )     (LDS)
// Phase 2: A = I - 0.99 * dinv_i * exp(20 g_ij)[i!=j] * dinv_j ; append Y.
// One block (256 threads) per example. Pad rows/cols never touched.
// ---------------------------------------------------------------------------
__global__ void k_affinity(float* __restrict__ aug) {
  const int e = blockIdx.x;
  float* A = aug + (size_t)e * AUGH * AUGW;
  __shared__ float dinv[MT];
  const int t = threadIdx.x;
  for (int i = t; i < MT; i += blockDim.x) {
    float s = 0.0f;
    for (int j = 0; j < MT; ++j) {
      const float g = A[(size_t)i * AUGW + j];
      float w = __expf(LS_GAMMA * g);
      if (j == i) w = 0.0f;
      s += w;
    }
    dinv[i] = rsqrtf(s);
  }
  __syncthreads();
  for (int idx = t; idx < MT * NAUG; idx += blockDim.x) {
    const int i = idx / NAUG;
    const int j = idx % NAUG;
    float out;
    if (j < MT) {
      if (i == j) {
        out = 1.0f;
      } else {
        const float g = A[(size_t)i * AUGW + j];
        out = -LS_FACTOR * dinv[i] * dinv[j] * __expf(LS_GAMMA * g);
      }
    } else {
      const int n = j - MT;                 // one-hot support labels (class-major)
      out = (i < MS && (i / HW) == n) ? 1.0f : 0.0f;
    }
    A[(size_t)i * AUGW + j] = out;
  }
}

// ---------------------------------------------------------------------------
// Kernel 5: Gauss-Jordan solve of augmented [A | Y] (294 x 299), in place.
// A = I - 0.99*Wnorm is SPD with unit diagonal -> no pivoting needed.
// Pivot row (scaled) and pivot column staged in LDS each step.
// ---------------------------------------------------------------------------
__global__ void k_solve(float* __restrict__ aug) {
  const int e = blockIdx.x;
  float* A = aug + (size_t)e * AUGH * AUGW;
  __shared__ float piv[NAUG];
  __shared__ float colk[MT];
  __shared__ float s_r;
  const int t = threadIdx.x;               // 256 threads
  for (int k = 0; k < MT; ++k) {
    if (t == 0) s_r = 1.0f / A[(size_t)k * AUGW + k];
    __syncthreads();
    const float r = s_r;
    for (int j = t; j < NAUG; j += blockDim.x) piv[j] = A[(size_t)k * AUGW + j] * r;
    for (int i = t; i < MT; i += blockDim.x) colk[i] = A[(size_t)i * AUGW + k];
    __syncthreads();
    for (int idx = t; idx < MT * NAUG; idx += blockDim.x) {
      const int i = idx / NAUG;
      const int j = idx % NAUG;
      const float pv = piv[j];
      if (i == k) A[(size_t)k * AUGW + j] = pv;
      else A[(size_t)i * AUGW + j] = fmaf(-colk[i], pv, A[(size_t)i * AUGW + j]);
    }
    __syncthreads();
  }
}

// ---------------------------------------------------------------------------
// Kernel 6: per-example loss. Fixed-order serial sums -> deterministic.
// ex[n] = mean over query rows 245..293 of F[:,n]; loss = -log(ex[y]/sum ex).
// ---------------------------------------------------------------------------
__global__ void k_loss(const float* __restrict__ aug,
                       const int* __restrict__ qy,
                       float* __restrict__ losses) {
  const int e = blockIdx.x;
  const float* A = aug + (size_t)e * AUGH * AUGW;
  __shared__ float ex[NW];
  const int t = threadIdx.x;               // 32 threads
  if (t < NW) {
    float s = 0.0f;
    for (int i = 0; i < HW; ++i) s += A[(size_t)(MS + i) * AUGW + (MT + t)];
    ex[t] = s * (1.0f / HW);
  }
  __syncthreads();
  if (t == 0) {
    float tot = 0.0f;
    for (int n = 0; n < NW; ++n) tot += ex[n];
    const int y = qy[e];
    losses[e] = -(logf(ex[y]) - logf(tot));
  }
}

// ---------------------------------------------------------------------------
// Kernel 7: deterministic mean of per-example losses -> d_out[0].
// ---------------------------------------------------------------------------
__global__ void k_reduce(const float* __restrict__ losses,
                         float* __restrict__ out, int nex) {
  __shared__ float red[256];
  const int t = threadIdx.x;
  float s = 0.0f;
  for (int i = t; i < nex; i += 256) s += losses[i];   // fixed strided order
  red[t] = s;
  __syncthreads();
  for (int o = 128; o > 0; o >>= 1) {
    if (t < o) red[t] += red[t + o];
    __syncthreads();
  }
  if (t == 0) out[0] = red[0] / (float)nex;
}

// ---------------------------------------------------------------------------
// Host launcher. Workspace layout (floats):
//   sflat : b*MS*C            (~2.5 MB)
//   qfn   : nex*HW*C          (~37.6 MB)
//   zcol  : C                 (2.5 KB, zero pad column)
//   aug   : nex*AUGH*AUGW     (~110.9 MB, 304x304 padded slabs)
//   losses: nex               (1.2 KB)
// Total ~151 MB of d_ws.
// ---------------------------------------------------------------------------
extern "C" void kernel_launch(void* const* d_in, const int* in_sizes, int n_in,
                              void* d_out, int out_size, void* d_ws, size_t ws_size,
                              hipStream_t stream) {
  (void)n_in; (void)out_size; (void)ws_size;
  const float* sup = (const float*)d_in[0];   // support_xf (b,25,640,7,7) f32
  const float* qry = (const float*)d_in[2];   // query_xf   (b,75,640,7,7) f32
  const int*   qy  = (const int*)d_in[3];     // query_y    (b,75)
  const int b   = in_sizes[0] / (S_TOT * C_DIM * HW);
  const int q   = in_sizes[2] / (b * C_DIM * HW);
  const int nex = b * q;

  float* ws     = (float*)d_ws;
  float* sflat  = ws;
  float* qfn    = sflat + (size_t)b * MS * C_DIM;
  float* zcol   = qfn + (size_t)nex * HW * C_DIM;
  float* aug    = zcol + C_DIM;
  float* losses = aug + (size_t)nex * AUGH * AUGW;
  float* out    = (float*)d_out;

  hipLaunchKernelGGL(k_zero, dim3(C_DIM / 64), dim3(64), 0, stream, zcol);
  hipLaunchKernelGGL(k_proto_norm, dim3(b * MS), dim3(64), 0, stream, sup, sflat);
  hipLaunchKernelGGL(k_query_norm, dim3(nex * HW), dim3(64), 0, stream, qry, qfn);
  hipLaunchKernelGGL(k_gram_wmma, dim3(nex * NTIL * NTIL), dim3(32), 0, stream,
                     sflat, qfn, zcol, aug, q);
  hipLaunchKernelGGL(k_affinity, dim3(nex), dim3(256), 0, stream, aug);
  hipLaunchKernelGGL(k_solve, dim3(nex), dim3(256), 0, stream, aug);
  hipLaunchKernelGGL(k_loss, dim3(nex), dim3(32), 0, stream, aug, qy, losses);
  hipLaunchKernelGGL(k_reduce, dim3(1), dim3(256), 0, stream, losses, out, nex);
}